// PhobiaLoss_11622181503749
// MI455X (gfx1250) — compile-verified
//
#include <hip/hip_runtime.h>
#include <cstdint>

typedef __attribute__((ext_vector_type(2))) float v2f;
typedef __attribute__((ext_vector_type(8))) float v8f;

#define EPSV 1e-6f

__device__ __forceinline__ float softplusf(float z) {
  return fmaxf(z, 0.f) + __logf(1.f + __expf(-fabsf(z)));
}

// Kernel 1: per-wave async-DMA staging of 32-cell tiles into LDS, per-cell loss
// terms, wave32 butterfly reduction, per-block partials (16 floats, cols 6..15 = 0).
__global__ void __launch_bounds__(256)
phobia_main(const float* __restrict__ pred, const float* __restrict__ tgt,
            float* __restrict__ partials, int n_cells) {
  __shared__ __align__(16) float stage[8 * 832];   // per wave: 416 pred + 416 tgt dwords
  __shared__ float blockred[8 * 8];

  const int lane = threadIdx.x & 31;
  const int wave = threadIdx.x >> 5;
  const uint32_t lds_base = (uint32_t)(uintptr_t)&stage[wave * 832];
  const uint64_t pbase = (uint64_t)(uintptr_t)pred;
  const uint64_t tbase = (uint64_t)(uintptr_t)tgt;
  const int wavesTotal = gridDim.x * (blockDim.x >> 5);
  const int gwave = blockIdx.x * (blockDim.x >> 5) + wave;
  const int nTiles = (n_cells + 31) >> 5;

  float aCoord = 0.f, aObj = 0.f, aNoobj = 0.f, aClass = 0.f, aNobj = 0.f, aNnoobj = 0.f;

  for (int tile = gwave; tile < nTiles; tile += wavesTotal) {
    const int cell0 = tile << 5;
    const int remCells = min(32, n_cells - cell0);
    const int nDw = remCells * 13;           // dwords per array this tile
    const int n16 = (nDw + 3) >> 2;          // 16-byte chunks (<= 104)
    const uint32_t gbyte = (uint32_t)cell0 * 52u;  // 16B-aligned tile base

    // make sure previous tile's LDS reads are fully drained before overwrite
    asm volatile("s_wait_dscnt 0" ::: "memory");

    #pragma unroll
    for (int k = 0; k < 4; ++k) {
      int idx = k * 32 + lane;
      if (idx < n16) {
        uint32_t go = gbyte + (uint32_t)idx * 16u;
        uint32_t la = lds_base + (uint32_t)idx * 16u;
        asm volatile("global_load_async_to_lds_b128 %0, %1, %2 offset:0"
                     :: "v"(la), "v"(go), "s"(pbase) : "memory");
        uint32_t la2 = la + 1664u;
        asm volatile("global_load_async_to_lds_b128 %0, %1, %2 offset:0"
                     :: "v"(la2), "v"(go), "s"(tbase) : "memory");
      }
    }
    asm volatile("s_wait_asynccnt 0" ::: "memory");

    if (lane < remCells) {
      const int pb = wave * 832 + lane * 13;   // stride 13: conflict-free on 64 banks
      const int tb = pb + 416;
      float p0 = stage[pb + 0], p1 = stage[pb + 1], p2 = stage[pb + 2], p3 = stage[pb + 3];
      float p4 = stage[pb + 4], p9 = stage[pb + 9];
      float pA = stage[pb + 10], pB = stage[pb + 11], pC = stage[pb + 12];
      float t0 = stage[tb + 0], t1 = stage[tb + 1], t2 = stage[tb + 2], t3 = stage[tb + 3];
      float t4 = stage[tb + 4];
      float tA = stage[tb + 10], tB = stage[tb + 11], tC = stage[tb + 12];

      float obj = (t4 > 0.f) ? 1.f : 0.f;
      float noobj = 1.f - obj;

      // coord
      float px = 1.f / (1.f + __expf(-p0));
      float py = 1.f / (1.f + __expf(-p1));
      float dx = px - t0, dy = py - t1;
      float sw = __fsqrt_rn(fabsf(p2) + EPSV) - __fsqrt_rn(fabsf(t2) + EPSV);
      float sh = __fsqrt_rn(fabsf(p3) + EPSV) - __fsqrt_rn(fabsf(t3) + EPSV);
      aCoord += obj * (dx * dx + dy * dy + sw * sw + sh * sh);

      // obj confidence BCE-with-logits
      float bce = fmaxf(p4, 0.f) - p4 * t4 + __logf(1.f + __expf(-fabsf(p4)));
      aObj += obj * bce;

      // no-obj softplus over both box logits
      aNoobj += noobj * (softplusf(p4) + softplusf(p9));

      // class CE: argmax(target) (first max), stable log-softmax
      float sel = pA; float best = tA;
      if (tB > best) { best = tB; sel = pB; }
      if (tC > best) { sel = pC; }
      float m = fmaxf(pA, fmaxf(pB, pC));
      float lse = m + __logf(__expf(pA - m) + __expf(pB - m) + __expf(pC - m));
      aClass += obj * (lse - sel);

      aNobj += obj;
      aNnoobj += noobj;
    }
  }

  // wave32 butterfly reduction
  #pragma unroll
  for (int off = 16; off >= 1; off >>= 1) {
    aCoord  += __shfl_xor(aCoord, off, 32);
    aObj    += __shfl_xor(aObj, off, 32);
    aNoobj  += __shfl_xor(aNoobj, off, 32);
    aClass  += __shfl_xor(aClass, off, 32);
    aNobj   += __shfl_xor(aNobj, off, 32);
    aNnoobj += __shfl_xor(aNnoobj, off, 32);
  }
  if (lane == 0) {
    blockred[wave * 8 + 0] = aCoord;
    blockred[wave * 8 + 1] = aObj;
    blockred[wave * 8 + 2] = aNoobj;
    blockred[wave * 8 + 3] = aClass;
    blockred[wave * 8 + 4] = aNobj;
    blockred[wave * 8 + 5] = aNnoobj;
  }
  __syncthreads();
  int t = threadIdx.x;
  if (t < 16) {
    float s = 0.f;
    if (t < 6) {
      #pragma unroll
      for (int w = 0; w < 8; ++w) s += blockred[w * 8 + t];
    }
    partials[blockIdx.x * 16 + t] = s;   // cols 6..15 written as 0
  }
}

// Kernel 2: fold nb x 16 partials to 16 x 16 (coalesced), then column-sum with
// 4 chained V_WMMA_F32_16X16X4_F32 (A = ones). D row 0 lanes 0..5 hold totals.
__global__ void __launch_bounds__(256)
phobia_final(const float* __restrict__ partials, float* __restrict__ out,
             int nb, float bsf) {
  __shared__ float mat[256];   // 16 x 16
  __shared__ float sums[16];
  int t = threadIdx.x;
  float acc = 0.f;
  int total = nb * 16;
  for (int i = t; i < total; i += 256) acc += partials[i];
  mat[t] = acc;
  __syncthreads();

  int lane = t & 31;
  int col = lane & 15;
  v2f a; a.x = 1.f; a.y = 1.f;          // A = ones(16x4)
  v8f c = {};
  #pragma unroll
  for (int g = 0; g < 4; ++g) {
    // B 4x16 layout: VGPR0 = rows K0 (lanes 0-15) / K2 (lanes 16-31), VGPR1 = K1/K3
    int r = g * 4 + ((lane >> 4) << 1);
    v2f b;
    b.x = mat[r * 16 + col];
    b.y = mat[(r + 1) * 16 + col];
    c = __builtin_amdgcn_wmma_f32_16x16x4_f32(false, a, false, b, (short)0, c,
                                              false, false);
  }
  if (t < 16) sums[t] = c[0];           // wave0, D row 0, cols 0..15
  __syncthreads();

  if (t == 0) {
    float coord = sums[0], objl = sums[1], noobjl = sums[2], classl = sums[3];
    float nObj = sums[4], nNoobj = sums[5];
    float cl = (nObj > 0.f) ? coord / bsf : coord;
    float ol = (nObj > 0.f) ? objl / bsf : objl;
    float kl = (nObj > 0.f) ? classl / bsf : classl;
    float nl = (nNoobj > 0.f) ? noobjl / bsf : noobjl;
    out[0] = 5.0f * cl + 5.0f * ol + 0.05f * nl + 1.0f * kl;
    out[1] = cl;
    out[2] = ol;
    out[3] = nl;
    out[4] = kl;
  }
}

extern "C" void kernel_launch(void* const* d_in, const int* in_sizes, int n_in,
                              void* d_out, int out_size, void* d_ws, size_t ws_size,
                              hipStream_t stream) {
  const float* pred = (const float*)d_in[0];
  const float* tgt  = (const float*)d_in[1];
  int n_pred  = in_sizes[0];
  int n_cells = n_pred / 13;            // N*S*S
  float bsf   = (float)(n_cells / 169); // batch size N

  float* partials = (float*)d_ws;
  int nb = 1024;
  size_t need = (size_t)nb * 16 * sizeof(float);
  if (need > ws_size) nb = (int)(ws_size / (16 * sizeof(float)));
  if (nb < 1) nb = 1;

  phobia_main<<<nb, 256, 0, stream>>>(pred, tgt, partials, n_cells);
  phobia_final<<<1, 256, 0, stream>>>(partials, (float*)d_out, nb, bsf);
}